// GD_Block_62242666053729
// MI455X (gfx1250) — compile-verified
//
#include <hip/hip_runtime.h>

// CDNA5 / gfx1250, wave32.
typedef __attribute__((ext_vector_type(2))) float v2f;
typedef __attribute__((ext_vector_type(8))) float v8f;

#define DCH 64   // channels

// -------------------------------------------------------------------------
// Hardware f32 atomic-add (non-returning, device scope) — guaranteed
// global_atomic_add_f32, never a CAS-loop expansion. One address pair,
// four adds via immediate offsets. Tracked on STOREcnt; implicit
// S_WAIT_IDLE at S_ENDPGM covers completion.
// -------------------------------------------------------------------------
__device__ __forceinline__ void atomic_add_f32x4(float* p, float4 v) {
    asm volatile(
        "global_atomic_add_f32 %0, %1, off scope:SCOPE_DEV\n\t"
        "global_atomic_add_f32 %0, %2, off offset:4 scope:SCOPE_DEV\n\t"
        "global_atomic_add_f32 %0, %3, off offset:8 scope:SCOPE_DEV\n\t"
        "global_atomic_add_f32 %0, %4, off offset:12 scope:SCOPE_DEV"
        :: "v"(p), "v"(v.x), "v"(v.y), "v"(v.z), "v"(v.w)
        : "memory");
}

__device__ __forceinline__ v2f ldv2(const float* p) { return *(const v2f*)p; }

// -------------------------------------------------------------------------
// zero the agg workspace
// -------------------------------------------------------------------------
__global__ void zero_kernel(float4* __restrict__ p, long long n4) {
    long long i = (long long)blockIdx.x * blockDim.x + threadIdx.x;
    if (i < n4) p[i] = make_float4(0.f, 0.f, 0.f, 0.f);
}

// -------------------------------------------------------------------------
// agg[dst] += x[src]   (segment_sum over edges), 16 lanes per edge
// -------------------------------------------------------------------------
__global__ void scatter_add_kernel(const float* __restrict__ x,
                                   const int* __restrict__ src,
                                   const int* __restrict__ dst,
                                   float* __restrict__ agg, int nE) {
    int t = blockIdx.x * blockDim.x + threadIdx.x;
    int e = t >> 4;
    if (e >= nE) return;
    int j = (t & 15) << 2;
    int s = src[e], d = dst[e];
    float4 v = *(const float4*)(x + (long long)s * DCH + j);
    atomic_add_f32x4(agg + (long long)d * DCH + j, v);
}

// -------------------------------------------------------------------------
// Fused 5-way GEMM with V_WMMA_F32_16X16X4_F32 (full fp32 precision):
//   out = x@W0^T + agg@W1^T ;  Q = x@WQ^T ;  K = x@WK^T ;  V = x@WV^T
// One wave per 16-row strip. A-fragment layout (32-bit 16x4 MxK):
//   lanes 0-15: M=lane, K={0,1}; lanes 16-31: M=lane-16, K={2,3}
// B-fragment (4x16 KxN) mirrors it with N across lanes; B[k][n] = W[n][k].
// C/D tile: VGPR r -> M = r + 8*(lane>=16), N = lane&15.
// -------------------------------------------------------------------------
__global__ void __launch_bounds__(128)
gemm5_kernel(const float* __restrict__ x,  const float* __restrict__ agg,
             const float* __restrict__ W0, const float* __restrict__ W1,
             const float* __restrict__ WQ, const float* __restrict__ WK,
             const float* __restrict__ WV,
             float* __restrict__ out, float* __restrict__ Qo,
             float* __restrict__ Ko,  float* __restrict__ Vo, int nrows)
{
    const int lane = threadIdx.x & 31;
    const int wave = threadIdx.x >> 5;
    const int row0 = (blockIdx.x * 4 + wave) * 16;   // wave-uniform guard: EXEC stays all-1s
    if (row0 >= nrows) return;
    const int m = lane & 15;
    const int h = lane >> 4;

    // Preload A fragments for x and agg: 16 K-steps of v2f each.
    v2f ax[16], ag[16];
    const float* xr = x   + (long long)(row0 + m) * DCH + 2 * h;
    const float* gr = agg + (long long)(row0 + m) * DCH + 2 * h;
#pragma unroll
    for (int kk = 0; kk < 16; ++kk) {
        ax[kk] = ldv2(xr + 4 * kk);
        ag[kk] = ldv2(gr + 4 * kk);
    }

    const int bi = m * DCH + 2 * h;   // per-lane base into W (row n0+m, col pair at 2h)

#pragma unroll
    for (int n0 = 0; n0 < DCH; n0 += 16) {
        const float* b0 = W0 + n0 * DCH + bi;
        const float* b1 = W1 + n0 * DCH + bi;
        const float* bq = WQ + n0 * DCH + bi;
        const float* bk = WK + n0 * DCH + bi;
        const float* bv = WV + n0 * DCH + bi;

        // ---- out tile: x*W0^T + agg*W1^T (32 chained WMMAs) ----
        v8f acc = {};
#pragma unroll
        for (int kk = 0; kk < 16; ++kk)
            acc = __builtin_amdgcn_wmma_f32_16x16x4_f32(
                false, ax[kk], false, ldv2(b0 + 4 * kk), (short)0, acc, false, false);
#pragma unroll
        for (int kk = 0; kk < 16; ++kk)
            acc = __builtin_amdgcn_wmma_f32_16x16x4_f32(
                false, ag[kk], false, ldv2(b1 + 4 * kk), (short)0, acc, false, false);
#pragma unroll
        for (int r = 0; r < 8; ++r)
            out[(long long)(row0 + r + 8 * h) * DCH + n0 + m] = acc[r];

        // ---- Q tile ----
        acc = (v8f){};
#pragma unroll
        for (int kk = 0; kk < 16; ++kk)
            acc = __builtin_amdgcn_wmma_f32_16x16x4_f32(
                false, ax[kk], false, ldv2(bq + 4 * kk), (short)0, acc, false, false);
#pragma unroll
        for (int r = 0; r < 8; ++r)
            Qo[(long long)(row0 + r + 8 * h) * DCH + n0 + m] = acc[r];

        // ---- K tile ----
        acc = (v8f){};
#pragma unroll
        for (int kk = 0; kk < 16; ++kk)
            acc = __builtin_amdgcn_wmma_f32_16x16x4_f32(
                false, ax[kk], false, ldv2(bk + 4 * kk), (short)0, acc, false, false);
#pragma unroll
        for (int r = 0; r < 8; ++r)
            Ko[(long long)(row0 + r + 8 * h) * DCH + n0 + m] = acc[r];

        // ---- V tile ----
        acc = (v8f){};
#pragma unroll
        for (int kk = 0; kk < 16; ++kk)
            acc = __builtin_amdgcn_wmma_f32_16x16x4_f32(
                false, ax[kk], false, ldv2(bv + 4 * kk), (short)0, acc, false, false);
#pragma unroll
        for (int r = 0; r < 8; ++r)
            Vo[(long long)(row0 + r + 8 * h) * DCH + n0 + m] = acc[r];
    }
}

// -------------------------------------------------------------------------
// Per edge e: score = <Q[d], K[s]>;  out[d] -= score * V[s]
// 16 lanes per edge, float4 per lane, shfl-xor tree over the 16 lanes.
// -------------------------------------------------------------------------
__global__ void attn_kernel(const float* __restrict__ Q, const float* __restrict__ K,
                            const float* __restrict__ V,
                            const int* __restrict__ s2, const int* __restrict__ d2,
                            float* __restrict__ out, int nE)
{
    int t = blockIdx.x * blockDim.x + threadIdx.x;
    int e = t >> 4;
    if (e >= nE) return;
    int j = (t & 15) << 2;
    int s = s2[e], d = d2[e];
    const float4 q = *(const float4*)(Q + (long long)d * DCH + j);
    const float4 k = *(const float4*)(K + (long long)s * DCH + j);
    float p = q.x * k.x + q.y * k.y + q.z * k.z + q.w * k.w;
#pragma unroll
    for (int off = 8; off >= 1; off >>= 1)
        p += __shfl_xor(p, off, 16);
    const float4 v = *(const float4*)(V + (long long)s * DCH + j);
    atomic_add_f32x4(out + (long long)d * DCH + j,
                     make_float4(-p * v.x, -p * v.y, -p * v.z, -p * v.w));
}

// -------------------------------------------------------------------------
extern "C" void kernel_launch(void* const* d_in, const int* in_sizes, int n_in,
                              void* d_out, int out_size, void* d_ws, size_t ws_size,
                              hipStream_t stream)
{
    const float* x  = (const float*)d_in[0];
    const int*  ei1 = (const int*)d_in[1];
    const int*  ei2 = (const int*)d_in[2];
    const float* W0 = (const float*)d_in[3];
    const float* W1 = (const float*)d_in[4];
    const float* WQ = (const float*)d_in[5];
    const float* WK = (const float*)d_in[6];
    const float* WV = (const float*)d_in[7];
    float* out = (float*)d_out;

    const int N  = in_sizes[0] / DCH;
    const int E1 = in_sizes[1] / 2;
    const int E2 = in_sizes[2] / 2;

    float* agg = (float*)d_ws;
    float* Qb  = agg + (size_t)N * DCH;
    float* Kb  = Qb  + (size_t)N * DCH;
    float* Vb  = Kb  + (size_t)N * DCH;

    // 1) zero agg
    long long n4 = (long long)N * DCH / 4;
    zero_kernel<<<(unsigned)((n4 + 255) / 256), 256, 0, stream>>>((float4*)agg, n4);

    // 2) agg = segment_sum(x[src], dst)
    {
        long long threads = (long long)E1 * 16;
        scatter_add_kernel<<<(unsigned)((threads + 255) / 256), 256, 0, stream>>>(
            x, ei1, ei1 + E1, agg, E1);
    }

    // 3) fused 5-GEMM (WMMA fp32): out = x@W0^T + agg@W1^T; Q,K,V projections
    {
        int strips = (N + 15) / 16;
        int blocks = (strips + 3) / 4;
        gemm5_kernel<<<blocks, 128, 0, stream>>>(x, agg, W0, W1, WQ, WK, WV,
                                                 out, Qb, Kb, Vb, N);
    }

    // 4) out[d] -= <Q[d],K[s]> * V[s] over edge set 2
    {
        long long threads = (long long)E2 * 16;
        attn_kernel<<<(unsigned)((threads + 255) / 256), 256, 0, stream>>>(
            Qb, Kb, Vb, ei2, ei2 + E2, out, E2);
    }
}